// CTCLoss_67869073211888
// MI455X (gfx1250) — compile-verified
//
#include <hip/hip_runtime.h>
#include <cstdint>

// Problem dimensions (fixed by the reference setup)
#define T_DIM 2048
#define N_DIM 128
#define C_DIM 96
#define L_DIM 200
#define S_DIM 401          // 2L+1 extended states
#define BLK   128          // 4 wave32 per workgroup
#define RPT   4            // extended states per thread (128*4 = 512 >= 401)
#define CH    32           // time steps staged per TDM chunk (T_DIM % CH == 0)

#define NEGF  (-1.0e30f)
#define LOG2E 1.4426950408889634f
#define LN2   0.6931471805599453f

typedef uint32_t u32x4 __attribute__((ext_vector_type(4)));
typedef uint32_t u32x8 __attribute__((ext_vector_type(8)));

// v_log_f32 computes log2, v_exp_f32 computes exp2 — recurrence in log2 domain.
__device__ __forceinline__ float lae3(float a, float b, float c) {
    float m = fmaxf(fmaxf(a, b), c);
    float s = __builtin_amdgcn_exp2f(a - m) +
              __builtin_amdgcn_exp2f(b - m) +
              __builtin_amdgcn_exp2f(c - m);
    return m + __builtin_amdgcn_logf(s);
}

// Tensor Data Mover: DMA a 2D tile (C_DIM floats x CH rows, row stride N*C
// elements) from global memory into LDS.  Descriptor bitfields per CDNA5 ISA
// (CDNA5 ISA): group0 = {count, lds_addr, global_addr, type=2},
// group1 = {data_size=4B, tensor_dim0=96, tensor_dim1=CH, tile 96xCH, stride}.
__device__ __forceinline__ void tdm_load_chunk(uint64_t ga, uint32_t lds_off) {
    u32x4 g0;
    g0.x = 0x1u;                                            // count=1, no gather
    g0.y = lds_off;                                         // LDS byte address
    g0.z = (uint32_t)ga;                                    // global addr [31:0]
    g0.w = (uint32_t)((ga >> 32) & 0x1FFFFFFu) | (2u << 30);// [56:32] | type=2
    u32x8 g1;
    g1[0] = 0x00020000u;                   // workgroup_mask=0, data_size=2 (4B)
    g1[1] = (uint32_t)C_DIM << 16;         // tensor_dim0[15:0] = 96
    g1[2] = (uint32_t)CH << 16;            // tensor_dim1[15:0] = CH
    g1[3] = (uint32_t)C_DIM << 16;         // tile_dim0 = 96
    g1[4] = (uint32_t)CH;                  // tile_dim1 = CH, tile_dim2 = 0
    g1[5] = (uint32_t)(N_DIM * C_DIM);     // tensor_dim0_stride[31:0] = 12288
    g1[6] = 0u;                            // stride0[47:32]=0, stride1[15:0]=0
    g1[7] = 0u;                            // stride1[47:16]=0
    asm volatile("tensor_load_to_lds %0, %1" :: "s"(g0), "s"(g1) : "memory");
}

__global__ void __launch_bounds__(BLK)
ctc_alpha_kernel(const float* __restrict__ lp,      // (T, N, C) log-probs
                 const int*   __restrict__ labels,  // (N, L)
                 const int*   __restrict__ tlen,    // (N,)
                 const int*   __restrict__ llen,    // (N,)
                 float* __restrict__ loss_out)      // (N,) -> d_ws
{
    __shared__ float  s_chunk[2][CH][C_DIM];   // ping-pong TDM-staged rows (24 KB)
    __shared__ float2 s_bnd[2][BLK + 1];       // boundary (a2,a3); slot 0 = NEG pad
    __shared__ float  s_alpha[BLK * RPT];      // final alpha dump

    const int n   = blockIdx.x;
    const int tid = threadIdx.x;
    const int Tn  = tlen[n];
    const int Ln  = llen[n];
    const int s0  = tid * RPT;

    // ext[s] = 0 for even s (blank); labels[(s-1)/2] for odd s. Only r=1,3 odd.
    const int* lrow = labels + n * L_DIM;
    int  e1 = 0, e3 = 0;
    bool sk1 = false, sk3 = false;
    {
        int s1 = s0 + 1, s3 = s0 + 3;
        if (s1 < S_DIM) {
            e1 = lrow[s1 >> 1];
            if (s1 >= 3) sk1 = (e1 != lrow[(s1 >> 1) - 1]);
        }
        if (s3 < S_DIM) {
            e3  = lrow[s3 >> 1];
            sk3 = (e3 != lrow[(s3 >> 1) - 1]);   // s3 >= 3 always
        }
    }

    // virtual alpha_{-1}: 0 at s=0 so the t=0 step yields the standard init
    float a0 = (s0 == 0) ? 0.0f : NEGF;
    float a1 = NEGF, a2 = NEGF, a3 = NEGF;
    s_bnd[0][tid + 1] = make_float2(a2, a3);
    s_bnd[1][tid + 1] = make_float2(NEGF, NEGF);
    if (tid == 0) {                       // NEG pad so the loop has no select
        s_bnd[0][0] = make_float2(NEGF, NEGF);
        s_bnd[1][0] = make_float2(NEGF, NEGF);
    }

    const uint64_t gbase = (uint64_t)(uintptr_t)(lp + (size_t)n * C_DIM);
    const uint32_t lds_c0 = (uint32_t)(uintptr_t)&s_chunk[0][0][0];
    const uint32_t lds_c1 = (uint32_t)(uintptr_t)&s_chunk[1][0][0];
    const uint64_t chunk_bytes = (uint64_t)CH * N_DIM * C_DIM * sizeof(float);

    // Kick off chunk 0 (wave 0 issues; TDM ignores EXEC, in-order per wave)
    if (tid < 32) tdm_load_chunk(gbase, lds_c0);

    const int nc = (Tn + CH - 1) / CH;
    for (int c = 0; c < nc; ++c) {
        // Prefetch chunk c+1 into the other buffer, then wait for chunk c.
        if (tid < 32) {
            if ((c + 1) * CH < T_DIM) {
                tdm_load_chunk(gbase + (uint64_t)(c + 1) * chunk_bytes,
                               (c & 1) ? lds_c0 : lds_c1);
                __builtin_amdgcn_s_wait_tensorcnt(1);   // chunk c landed
            } else {
                __builtin_amdgcn_s_wait_tensorcnt(0);
            }
        }
        __syncthreads();

        const int tmax = min(CH, Tn - c * CH);
        const float (*lprow)[C_DIM] = s_chunk[c & 1];
        for (int tin = 0; tin < tmax; ++tin) {
            const int cur = tin & 1;   // CH even => parity continuous across chunks

            // Gather log2-domain log-probs (even states are blank = class 0)
            float lpb  = lprow[tin][0]  * LOG2E;
            float lpe1 = lprow[tin][e1] * LOG2E;
            float lpe3 = lprow[tin][e3] * LOG2E;

            // Boundary alphas from previous thread (slot 0 is the NEG pad)
            float2 pb = s_bnd[cur][tid];
            float p2 = pb.x, p3 = pb.y;

            float n0 = lae3(a0, p3, NEGF)            + lpb;   // even s: no skip
            float n1 = lae3(a1, a0, sk1 ? p3 : NEGF) + lpe1;
            float n2 = lae3(a2, a1, NEGF)            + lpb;   // even s: no skip
            float n3 = lae3(a3, a2, sk3 ? a1 : NEGF) + lpe3;
            a0 = n0; a1 = n1; a2 = n2; a3 = n3;

            s_bnd[cur ^ 1][tid + 1] = make_float2(a2, a3);
            __syncthreads();
        }
    }

    // Loop stopped at t = Tn, so regs hold alpha after step Tn-1 (capture point).
    s_alpha[s0 + 0] = a0; s_alpha[s0 + 1] = a1;
    s_alpha[s0 + 2] = a2; s_alpha[s0 + 3] = a3;
    __syncthreads();

    if (tid == 0) {
        int   end = 2 * Ln;                 // trailing blank position
        float fa = s_alpha[end], fb = s_alpha[end - 1];
        float m  = fmaxf(fa, fb);
        float f2 = m + __builtin_amdgcn_logf(__builtin_amdgcn_exp2f(fa - m) +
                                             __builtin_amdgcn_exp2f(fb - m));
        loss_out[n] = -f2 * LN2;            // back to natural-log domain
    }
}

// focal reweighting + deterministic tree-sum: sum_n loss * (1 - exp(-loss))^2
__global__ void __launch_bounds__(N_DIM)
focal_reduce(const float* __restrict__ pl, float* __restrict__ out)
{
    __shared__ float red[N_DIM];
    int t = threadIdx.x;
    float loss = pl[t];
    float e  = __builtin_amdgcn_exp2f(-loss * LOG2E);   // exp(-loss)
    float om = 1.0f - e;
    red[t] = loss * om * om;                            // ALPHA=1, GAMMA=2
    __syncthreads();
    for (int off = N_DIM / 2; off > 0; off >>= 1) {
        if (t < off) red[t] += red[t + off];
        __syncthreads();
    }
    if (t == 0) out[0] = red[0];
}

extern "C" void kernel_launch(void* const* d_in, const int* in_sizes, int n_in,
                              void* d_out, int out_size, void* d_ws, size_t ws_size,
                              hipStream_t stream) {
    const float* predicts = (const float*)d_in[0];  // (T,N,C) f32
    const int*   labels   = (const int*)d_in[1];    // (N,L) i32
    const int*   tlen     = (const int*)d_in[3];    // (N,) i32
    const int*   llen     = (const int*)d_in[4];    // (N,) i32
    float* per_batch = (float*)d_ws;                // N floats of scratch

    ctc_alpha_kernel<<<dim3(N_DIM), dim3(BLK), 0, stream>>>(
        predicts, labels, tlen, llen, per_batch);
    focal_reduce<<<dim3(1), dim3(N_DIM), 0, stream>>>(per_batch, (float*)d_out);
}